// DistSAGE_55027120997010
// MI455X (gfx1250) — compile-verified
//
#include <hip/hip_runtime.h>

// ---------------------------------------------------------------------------
// DistSAGE on MI455X (gfx1250): edge scatter (fp32 atomics) + fused
// dual-GEMM per layer using V_WMMA_F32_16X16X4_F32 (exact fp32 matrix math).
// Round 2: interleaved dual accumulators -> two independent WMMA chains.
// ---------------------------------------------------------------------------

typedef __attribute__((ext_vector_type(2))) float v2f;
typedef __attribute__((ext_vector_type(8))) float v8f;

#define NS0 1081344
#define ND0 67584
#define NE0 1013760
#define ND1 6144
#define NE1 61440
#define ND2 1024
#define NE2 5120
#define INC 128
#define HID 256
#define OUTC 64

#define GEMM_THREADS 128  // 4 wave32 waves -> 16x64 output strip per block

// ---------------------------------------------------------------------------
// Edge scatter: msg[ed[e]] += H[es[e]]  (float4 per thread, fp32 atomics)
// C=128 -> one wave == one edge (512B coalesced); C=256 -> two waves/edge.
// ---------------------------------------------------------------------------
__global__ void sage_scatter(const float* __restrict__ H,
                             const int* __restrict__ es,
                             const int* __restrict__ ed,
                             float* __restrict__ msg,
                             int ne, int C)
{
    const int vecC = C >> 2;
    const long long total = (long long)ne * vecC;
    long long t = (long long)blockIdx.x * blockDim.x + threadIdx.x;
    if (t >= total) return;
    int e  = (int)(t / vecC);
    int c4 = (int)(t - (long long)e * vecC);
    int s = es[e], d = ed[e];
    const float4 v = ((const float4*)(H + (size_t)s * C))[c4];
    float* mp = msg + (size_t)d * C + (size_t)c4 * 4;
    atomicAdd(mp + 0, v.x);
    atomicAdd(mp + 1, v.y);
    atomicAdd(mp + 2, v.z);
    atomicAdd(mp + 3, v.w);
}

__global__ void sage_degree(const int* __restrict__ ed,
                            float* __restrict__ deg, int ne)
{
    int t = blockIdx.x * blockDim.x + threadIdx.x;
    if (t < ne) atomicAdd(&deg[ed[t]], 1.0f);
}

// msg[i, :] /= max(deg[i], 1)   (in place -> becomes the mean)
__global__ void sage_mean(float* __restrict__ msg,
                          const float* __restrict__ deg,
                          int n, int C)
{
    long long t = (long long)blockIdx.x * blockDim.x + threadIdx.x;
    long long total = (long long)n * C;
    if (t >= total) return;
    int i = (int)(t / C);
    msg[t] = msg[t] / fmaxf(deg[i], 1.0f);
}

// ---------------------------------------------------------------------------
// Fused dual GEMM + bias (+ReLU):
//   out[mb:mb+16, nb:nb+16] = Hself_tile @ Wself + Hmean_tile @ Wneigh + b
// A tiles staged in padded LDS; one 16x16 tile per wave. Single K-loop with
// two independent V_WMMA_F32_16X16X4_F32 accumulator chains (self / neigh),
// combined in the epilogue.
// ---------------------------------------------------------------------------
__global__ __launch_bounds__(GEMM_THREADS)
void sage_gemm_wmma(const float* __restrict__ Hself,
                    const float* __restrict__ Hmean,
                    const float* __restrict__ Wself,
                    const float* __restrict__ Wneigh,
                    const float* __restrict__ bias,
                    float* __restrict__ out,
                    int din, int dout, int do_relu)
{
    extern __shared__ float lds[];
    const int lstride = din + 4;          // +4 floats: conflict-free lane->bank map
    float* As = lds;                      // 16 x din  (Hself tile)
    float* Am = lds + 16 * lstride;       // 16 x din  (Hmean tile)

    const int m_base = blockIdx.x * 16;
    const int tid = threadIdx.x;

    // Cooperative stage of both A tiles (coalesced along rows).
    for (int i = tid; i < 16 * din; i += GEMM_THREADS) {
        int r = i / din;
        int c = i - r * din;
        As[r * lstride + c] = Hself[(size_t)(m_base + r) * din + c];
        Am[r * lstride + c] = Hmean[(size_t)(m_base + r) * din + c];
    }
    __syncthreads();

    const int wave = tid >> 5;
    const int lane = tid & 31;
    const int half = lane >> 4;           // 0: lanes 0-15, 1: lanes 16-31
    const int lm   = lane & 15;
    const int n_base = blockIdx.y * 64 + wave * 16;
    if (n_base >= dout) return;           // wave-uniform: EXEC stays all-ones

    const float* arow_s = As + lm * lstride;
    const float* arow_m = Am + lm * lstride;
    const float* bcol_s = Wself  + n_base + lm;   // column n, row-strided by dout
    const float* bcol_n = Wneigh + n_base + lm;

    v8f acc_s = {};
    v8f acc_m = {};

    // Interleaved K-loop: two independent WMMA chains share the address math.
    for (int k = 0; k < din; k += 4) {
        const size_t ka = (size_t)(k + 2 * half);  // K split per ISA: {0,1 | 2,3}
        v2f a_s, a_m, b_s, b_n;
        a_s.x = arow_s[ka];
        a_s.y = arow_s[ka + 1];
        a_m.x = arow_m[ka];
        a_m.y = arow_m[ka + 1];
        b_s.x = bcol_s[ka * dout];
        b_s.y = bcol_s[(ka + 1) * dout];
        b_n.x = bcol_n[ka * dout];
        b_n.y = bcol_n[(ka + 1) * dout];
        acc_s = __builtin_amdgcn_wmma_f32_16x16x4_f32(
            /*neg_a=*/false, a_s, /*neg_b=*/false, b_s,
            /*c_mod=*/(short)0, acc_s, /*reuse_a=*/false, /*reuse_b=*/false);
        acc_m = __builtin_amdgcn_wmma_f32_16x16x4_f32(
            false, a_m, false, b_n, (short)0, acc_m, false, false);
    }

    // Epilogue: C/D layout -> lane (half,lm) holds rows r+8*half, col n_base+lm.
    const float bn = bias[n_base + lm];
#pragma unroll
    for (int r = 0; r < 8; ++r) {
        float v = acc_s[r] + acc_m[r] + bn;
        if (do_relu) v = fmaxf(v, 0.0f);
        out[(size_t)(m_base + r + 8 * half) * dout + n_base + lm] = v;
    }
}

// ---------------------------------------------------------------------------
// Host-side orchestration (graph-capture safe: only async ops on `stream`).
// ---------------------------------------------------------------------------
static inline int ceil_div_ll(long long a, int b) { return (int)((a + b - 1) / b); }

extern "C" void kernel_launch(void* const* d_in, const int* in_sizes, int n_in,
                              void* d_out, int out_size, void* d_ws, size_t ws_size,
                              hipStream_t stream)
{
    const float* x   = (const float*)d_in[0];
    const int* e0s   = (const int*)d_in[1];
    const int* e0d   = (const int*)d_in[2];
    const int* e1s   = (const int*)d_in[3];
    const int* e1d   = (const int*)d_in[4];
    const int* e2s   = (const int*)d_in[5];
    const int* e2d   = (const int*)d_in[6];
    const float* Ws0 = (const float*)d_in[7];
    const float* Wn0 = (const float*)d_in[8];
    const float* b0  = (const float*)d_in[9];
    const float* Ws1 = (const float*)d_in[10];
    const float* Wn1 = (const float*)d_in[11];
    const float* b1  = (const float*)d_in[12];
    const float* Ws2 = (const float*)d_in[13];
    const float* Wn2 = (const float*)d_in[14];
    const float* b2  = (const float*)d_in[15];
    float* outp      = (float*)d_out;

    // Workspace carve-up (floats, 256B-aligned regions).
    float* ws = (float*)d_ws;
    size_t off = 0;
    auto carve = [&](size_t nfloats) {
        float* p = ws + off;
        off += (nfloats + 63) & ~(size_t)63;
        return p;
    };
    float* h1   = carve((size_t)ND0 * HID);   // layer-0 output (relu'd)
    float* h2   = carve((size_t)ND1 * HID);   // layer-1 output (relu'd)
    float* msg0 = carve((size_t)ND0 * INC);
    float* msg1 = carve((size_t)ND1 * HID);
    float* msg2 = carve((size_t)ND2 * HID);
    float* deg0 = carve(ND0);
    float* deg1 = carve(ND1);
    float* deg2 = carve(ND2);
    (void)ws_size; (void)n_in; (void)in_sizes; (void)out_size;

    // ---------------- Layer 0: din=128 -> dout=256, ReLU ----------------
    hipMemsetAsync(msg0, 0, (size_t)ND0 * INC * sizeof(float), stream);
    hipMemsetAsync(deg0, 0, (size_t)ND0 * sizeof(float), stream);
    {
        long long tot = (long long)NE0 * (INC / 4);
        sage_scatter<<<ceil_div_ll(tot, 256), 256, 0, stream>>>(x, e0s, e0d, msg0, NE0, INC);
        sage_degree<<<ceil_div_ll(NE0, 256), 256, 0, stream>>>(e0d, deg0, NE0);
        sage_mean<<<ceil_div_ll((long long)ND0 * INC, 256), 256, 0, stream>>>(msg0, deg0, ND0, INC);
        dim3 grid(ND0 / 16, HID / 64);
        size_t ldsz = (size_t)2 * 16 * (INC + 4) * sizeof(float);
        sage_gemm_wmma<<<grid, GEMM_THREADS, ldsz, stream>>>(
            x, msg0, Ws0, Wn0, b0, h1, INC, HID, /*relu=*/1);
    }

    // ---------------- Layer 1: din=256 -> dout=256, ReLU ----------------
    hipMemsetAsync(msg1, 0, (size_t)ND1 * HID * sizeof(float), stream);
    hipMemsetAsync(deg1, 0, (size_t)ND1 * sizeof(float), stream);
    {
        long long tot = (long long)NE1 * (HID / 4);
        sage_scatter<<<ceil_div_ll(tot, 256), 256, 0, stream>>>(h1, e1s, e1d, msg1, NE1, HID);
        sage_degree<<<ceil_div_ll(NE1, 256), 256, 0, stream>>>(e1d, deg1, NE1);
        sage_mean<<<ceil_div_ll((long long)ND1 * HID, 256), 256, 0, stream>>>(msg1, deg1, ND1, HID);
        dim3 grid(ND1 / 16, HID / 64);
        size_t ldsz = (size_t)2 * 16 * (HID + 4) * sizeof(float);
        sage_gemm_wmma<<<grid, GEMM_THREADS, ldsz, stream>>>(
            h1, msg1, Ws1, Wn1, b1, h2, HID, HID, /*relu=*/1);
    }

    // ---------------- Layer 2: din=256 -> dout=64, no ReLU ----------------
    hipMemsetAsync(msg2, 0, (size_t)ND2 * HID * sizeof(float), stream);
    hipMemsetAsync(deg2, 0, (size_t)ND2 * sizeof(float), stream);
    {
        long long tot = (long long)NE2 * (HID / 4);
        sage_scatter<<<ceil_div_ll(tot, 256), 256, 0, stream>>>(h2, e2s, e2d, msg2, NE2, HID);
        sage_degree<<<ceil_div_ll(NE2, 256), 256, 0, stream>>>(e2d, deg2, NE2);
        sage_mean<<<ceil_div_ll((long long)ND2 * HID, 256), 256, 0, stream>>>(msg2, deg2, ND2, HID);
        dim3 grid(ND2 / 16, 1);  // dout=64 covered by 4 waves
        size_t ldsz = (size_t)2 * 16 * (HID + 4) * sizeof(float);
        sage_gemm_wmma<<<grid, GEMM_THREADS, ldsz, stream>>>(
            h2, msg2, Ws2, Wn2, b2, outp, HID, OUTC, /*relu=*/0);
    }
}